// PMLoss_13383118094344
// MI455X (gfx1250) — compile-verified
//
#include <hip/hip_runtime.h>

// ---------------------------------------------------------------------------
// PM/coherence-diffusion PDE loss for MI455X (gfx1250, wave32, WMMA).
// Separable Gaussian blurs run on the fp32 matrix pipe via
// V_WMMA_F32_16X16X4_F32 band-matrix products; band coefficients are
// regenerated arithmetically per lane (v_exp_f32 on the TRANS pipe) so the
// WMMA K-loop contains no divergent memory traffic. Everything else is fused
// stencil code staged through the 192MB L2.
// ---------------------------------------------------------------------------

typedef float v2f __attribute__((ext_vector_type(2)));
typedef float v8f __attribute__((ext_vector_type(8)));

#define IMG_H 1024
#define IMG_W 1024
#define HW (1u << 20)

// Gaussian band coefficient wt[d] = norm * exp(-(d-R)^2 / (2*sigma^2)),
// zero outside [0, TAPS). norm = 1/sum of unnormalized taps.
// TAPS=7: sigma=1.0 -> exponent scale (log2 e)/2 = 0.72134752
// TAPS=5: sigma=0.5 -> exponent scale 2*log2 e   = 2.88539008
template <int TAPS>
__device__ __forceinline__ float gaussw(int d) {
  constexpr float c2 = (TAPS == 7) ? 0.72134752f : 2.88539008f;
  constexpr float norm = (TAPS == 7) ? 0.39905028f : 0.78657075f;
  float dd = (float)(d - TAPS / 2);
  float w = norm * exp2f(-c2 * dd * dd);
  return ((unsigned)d < (unsigned)TAPS) ? w : 0.0f;
}

__device__ __forceinline__ float ldz(const float* p, int r, int c) {
  return (r >= 0 && r < IMG_H && c >= 0 && c < IMG_W) ? p[(r << 10) + c] : 0.0f;
}

// Sobel (correlation form) at (r,c) with zero padding.
__device__ __forceinline__ void sobel_at(const float* p, int r, int c,
                                         float& sx, float& sy) {
  float a00 = ldz(p, r - 1, c - 1), a01 = ldz(p, r - 1, c), a02 = ldz(p, r - 1, c + 1);
  float a10 = ldz(p, r, c - 1),                              a12 = ldz(p, r, c + 1);
  float a20 = ldz(p, r + 1, c - 1), a21 = ldz(p, r + 1, c), a22 = ldz(p, r + 1, c + 1);
  sx = (a02 - a00) + 2.0f * (a12 - a10) + (a22 - a20);
  sy = (a00 + 2.0f * a01 + a02) - (a20 + 2.0f * a21 + a22);
}

// ---------------------------------------------------------------------------
// Separable TAPS-tap blur of one 16x16 tile per wave using WMMA f32 16x16x4.
//   pass1 (vertical):   T[i][t] = sum_m A[i][m]*X[m][t],  A[i][m]=wt[m-i]
//   pass2 (horizontal): O[i][j] = sum_m T[i][m]*B[m][j],  B[m][j]=wt[m-j]
// The per-lane band fragments for pass-1 A and pass-2 B are identical
// (wt[m0-lid]); they are computed once into registers and reused.
// Fragment layouts per the CDNA5 ISA:
//   A 16x4 f32: lanes 0-15 hold K {4k,4k+1}, lanes 16-31 hold {4k+2,4k+3}
//   B 4x16 f32: row striped across lanes (same half-wave K split)
//   C/D:        vgpr v -> row v (lanes 0-15) / v+8 (lanes 16-31), col = lane&15
// ---------------------------------------------------------------------------
template <int TAPS>
__device__ __forceinline__ void wave_sepconv(const float (*X)[144], float (*Tw)[24],
                                             float* out, int r0, int c0, int bx,
                                             int lane) {
  constexpr int VALID = 16 + (TAPS - 1);   // 22 (7-tap) or 20 (5-tap)
  constexpr int KC = (VALID + 3) / 4;      // K chunks of 4
  const int half = (lane >= 16) ? 1 : 0;
  const int lid = lane & 15;

  // Per-lane band coefficients, shared by pass-1 A and pass-2 B fragments.
  float bw[2 * KC];
#pragma unroll
  for (int kk = 0; kk < KC; ++kk) {
    int m0 = 4 * kk + 2 * half;
    bw[2 * kk] = gaussw<TAPS>(m0 - lid);
    bw[2 * kk + 1] = gaussw<TAPS>(m0 + 1 - lid);
  }

  // ---- pass 1: vertical band-matrix * X, two 16-col chunks of T ----
#pragma unroll
  for (int cc = 0; cc < 2; ++cc) {
    v8f acc = {};
#pragma unroll
    for (int kk = 0; kk < KC; ++kk) {
      int m0 = 4 * kk + 2 * half;
      v2f a;
      a.x = bw[2 * kk];
      a.y = bw[2 * kk + 1];
      v2f b;
      int xc = bx + 16 * cc + lid;
      b.x = X[m0][xc];
      b.y = X[m0 + 1][xc];
      acc = __builtin_amdgcn_wmma_f32_16x16x4_f32(false, a, false, b, (short)0,
                                                  acc, false, false);
    }
    int t = 16 * cc + lid;
    if (t < 24) {
#pragma unroll
      for (int v = 0; v < 8; ++v)
        Tw[v + 8 * half][t] = (t < VALID) ? acc[v] : 0.0f;  // zero K pads
    }
  }
  __syncthreads();  // uniform; orders Tw stores vs. pass-2 loads

  // ---- pass 2: T * horizontal band matrix ----
  v8f acc = {};
#pragma unroll
  for (int kk = 0; kk < KC; ++kk) {
    int m0 = 4 * kk + 2 * half;
    v2f a;
    a.x = Tw[lid][m0];
    a.y = Tw[lid][m0 + 1];
    v2f b;
    b.x = bw[2 * kk];
    b.y = bw[2 * kk + 1];
    acc = __builtin_amdgcn_wmma_f32_16x16x4_f32(false, a, false, b, (short)0,
                                                acc, false, false);
  }
  int col = c0 + lid;
#pragma unroll
  for (int v = 0; v < 8; ++v)
    out[((size_t)(r0 + v + 8 * half) << 10) + col] = acc[v];
}

// ---------------------------------------------------------------------------
// Kernel 1: 7x7 Gaussian blur of the 16 input images (8 pred + 8 target).
// Block = 256 threads = 8 waves; block tile = 16 rows x 128 cols.
// ---------------------------------------------------------------------------
__global__ void blur7_kernel(const float* __restrict__ pred,
                             const float* __restrict__ targ,
                             float* __restrict__ isig) {
  __shared__ float X[24][144];
  __shared__ float T[8][16][24];
  const int tid = threadIdx.x, lane = tid & 31, w = tid >> 5;
  const int rb = blockIdx.y * 16, cb = blockIdx.x * 128;
  const int img = blockIdx.z;
  const float* src =
      (img < 8) ? pred + (size_t)img * HW : targ + (size_t)(img - 8) * HW;

  for (int i = tid; i < 24 * 144; i += 256) {
    int r = i / 144, c = i % 144;
    X[r][c] = ldz(src, rb - 3 + r, cb - 3 + c);
  }
  __syncthreads();
  wave_sepconv<7>(X, T[w], isig + (size_t)img * HW, rb, cb + 16 * w, 16 * w,
                  lane);
}

// ---------------------------------------------------------------------------
// Kernel 2: fused Sobel-square + 5x5 Gaussian blur (structure tensor).
// grid.z = 32: image = z>>1, channel = z&1 (0: Isx^2 -> J11, 1: Isy^2 -> J22).
// ---------------------------------------------------------------------------
__global__ void sobelsq_blur5_kernel(const float* __restrict__ isig,
                                     float* __restrict__ j11,
                                     float* __restrict__ j22) {
  __shared__ float I[22][136];
  __shared__ float X[20][144];
  __shared__ float T[8][16][24];
  const int tid = threadIdx.x, lane = tid & 31, w = tid >> 5;
  const int rb = blockIdx.y * 16, cb = blockIdx.x * 128;
  const int z = blockIdx.z, img = z >> 1, ch = z & 1;
  const float* src = isig + (size_t)img * HW;

  for (int i = tid; i < 22 * 136; i += 256) {
    int r = i / 136, c = i % 136;
    I[r][c] = ldz(src, rb - 3 + r, cb - 3 + c);
  }
  __syncthreads();
  for (int i = tid; i < 20 * 144; i += 256) {
    int r = i / 144, c = i % 144;
    float val = 0.0f;
    if (c < 132) {  // cols >=132 feed only discarded T columns; keep finite
      float a00 = I[r][c], a01 = I[r][c + 1], a02 = I[r][c + 2];
      float a10 = I[r + 1][c], a12 = I[r + 1][c + 2];
      float a20 = I[r + 2][c], a21 = I[r + 2][c + 1], a22 = I[r + 2][c + 2];
      float s = (ch == 0)
                    ? (a02 - a00) + 2.0f * (a12 - a10) + (a22 - a20)
                    : (a00 + 2.0f * a01 + a02) - (a20 + 2.0f * a21 + a22);
      val = s * s;
    }
    X[r][c] = val;
  }
  __syncthreads();
  float* out = ((ch ? j22 : j11) + (size_t)img * HW);
  wave_sepconv<5>(X, T[w], out, rb, cb + 16 * w, 16 * w, lane);
}

// ---------------------------------------------------------------------------
// Kernel 3: edge-stop maps.  edge_stop(sqrt(s+1e-6)) == 1/(1+(s+1e-6)/K^2).
// ---------------------------------------------------------------------------
__global__ void gmaps_kernel(const float* __restrict__ isig,
                             const float* __restrict__ j11,
                             const float* __restrict__ j22,
                             float* __restrict__ g_i, float* __restrict__ g_j) {
  const int img = blockIdx.y;
  const int p = blockIdx.x * 256 + threadIdx.x;
  const int r = p >> 10, c = p & 1023;
  const size_t o = (size_t)img * HW + p;
  float sx, sy;
  sobel_at(isig + (size_t)img * HW, r, c, sx, sy);
  float s2 = sx * sx + sy * sy;
  g_j[o] = 1.0f / (1.0f + (s2 + 1e-6f) * (1.0f / 400.0f));
  float jt = j11[o] + j22[o];
  g_i[o] = 1.0f / (1.0f + (jt + 1e-6f) * (1.0f / 400.0f));
}

// ---------------------------------------------------------------------------
// Divergence helpers for the loss kernels: stage flux = g * sobel(field) for
// an 18x18 halo tile in LDS, then apply Sobel again for the 16x16 centers.
// ---------------------------------------------------------------------------
__device__ __forceinline__ void stage_flux(const float* __restrict__ F,
                                           const float* __restrict__ G, int r0,
                                           int c0, int tid, float (*fx)[18],
                                           float (*fy)[18]) {
  for (int i = tid; i < 18 * 18; i += 256) {
    int qr = i / 18, qc = i % 18;
    int gr = r0 - 1 + qr, gc = c0 - 1 + qc;
    float vx = 0.0f, vy = 0.0f;
    if (gr >= 0 && gr < IMG_H && gc >= 0 && gc < IMG_W) {
      float g = G[(gr << 10) + gc];
      float sx, sy;
      sobel_at(F, gr, gc, sx, sy);
      vx = g * sx;
      vy = g * sy;
    }
    fx[qr][qc] = vx;
    fy[qr][qc] = vy;
  }
}

__device__ __forceinline__ float div_from_flux(const float (*fx)[18],
                                               const float (*fy)[18], int tr,
                                               int tc) {
  float dx = (fx[tr][tc + 2] - fx[tr][tc]) +
             2.0f * (fx[tr + 1][tc + 2] - fx[tr + 1][tc]) +
             (fx[tr + 2][tc + 2] - fx[tr + 2][tc]);
  float dy = (fy[tr][tc] + 2.0f * fy[tr][tc + 1] + fy[tr][tc + 2]) -
             (fy[tr + 2][tc] + 2.0f * fy[tr + 2][tc + 1] + fy[tr + 2][tc + 2]);
  return dx + dy;
}

__device__ __forceinline__ void block_reduce_add(float val, int tid,
                                                 float* red, float* acc) {
  red[tid] = val;
  __syncthreads();
  for (int s = 128; s > 0; s >>= 1) {
    if (tid < s) red[tid] += red[tid + s];
    __syncthreads();
  }
  if (tid == 0) atomicAdd(acc, red[0]);
}

// Kernel 4: sum of (div_I_pred - div_I_target)^2 over one 16x16 tile per block.
__global__ void loss_i_kernel(const float* __restrict__ pred,
                              const float* __restrict__ targ,
                              const float* __restrict__ g_i,
                              float* __restrict__ acc) {
  __shared__ float fx[18][18], fy[18][18], red[256];
  const int tid = threadIdx.x, tr = tid >> 4, tc = tid & 15;
  const int r0 = blockIdx.y * 16, c0 = blockIdx.x * 16, n = blockIdx.z;
  float d[2];
  for (int s = 0; s < 2; ++s) {
    const float* F = (s ? targ : pred) + (size_t)n * HW;
    const float* G = g_i + (size_t)(s ? n + 8 : n) * HW;
    stage_flux(F, G, r0, c0, tid, fx, fy);
    __syncthreads();
    d[s] = div_from_flux(fx, fy, tr, tc);
    __syncthreads();
  }
  float e = d[0] - d[1];
  block_reduce_add(e * e, tid, red, acc);
}

// Kernel 5: sum of (div_J_pred - div_J_target)^2; div_J = div(J11,gJ)+div(J22,gJ).
__global__ void loss_j_kernel(const float* __restrict__ j11,
                              const float* __restrict__ j22,
                              const float* __restrict__ g_j,
                              float* __restrict__ acc) {
  __shared__ float fx[18][18], fy[18][18], red[256];
  const int tid = threadIdx.x, tr = tid >> 4, tc = tid & 15;
  const int r0 = blockIdx.y * 16, c0 = blockIdx.x * 16, n = blockIdx.z;
  float d[2];
  for (int s = 0; s < 2; ++s) {
    const int img = s ? n + 8 : n;
    const float* G = g_j + (size_t)img * HW;
    float dsum = 0.0f;
    for (int ch = 0; ch < 2; ++ch) {
      const float* F = (ch ? j22 : j11) + (size_t)img * HW;
      stage_flux(F, G, r0, c0, tid, fx, fy);
      __syncthreads();
      dsum += div_from_flux(fx, fy, tr, tc);
      __syncthreads();
    }
    d[s] = dsum;
  }
  float e = d[0] - d[1];
  block_reduce_add(e * e, tid, red, acc);
}

__global__ void init_acc_kernel(float* acc) { acc[0] = 0.0f; }

__global__ void finalize_kernel(const float* __restrict__ acc,
                                float* __restrict__ out) {
  // loss_I and loss_J share the denominator 8*1024*1024, so one accumulator.
  out[0] = acc[0] * (1.0f / 8388608.0f);
}

// ---------------------------------------------------------------------------
extern "C" void kernel_launch(void* const* d_in, const int* in_sizes, int n_in,
                              void* d_out, int out_size, void* d_ws,
                              size_t ws_size, hipStream_t stream) {
  (void)in_sizes; (void)n_in; (void)out_size; (void)ws_size;
  const float* pred = (const float*)d_in[0];   // (8,1,1024,1024) f32
  const float* targ = (const float*)d_in[1];   // (8,1,1024,1024) f32
  float* ws = (float*)d_ws;
  // Scratch layout (floats): 16 images per array (0-7 pred, 8-15 target).
  float* isig = ws;                     // 16*HW : 7x7-blurred images
  float* j11 = ws + (size_t)16 * HW;    // 16*HW : G5 * Isx^2
  float* j22 = ws + (size_t)32 * HW;    // 16*HW : G5 * Isy^2
  float* g_i = ws + (size_t)48 * HW;    // 16*HW : edge-stop for div_I
  float* g_j = ws + (size_t)64 * HW;    // 16*HW : edge-stop for div_J
  float* acc = ws + (size_t)80 * HW;    // 1     : global loss accumulator

  blur7_kernel<<<dim3(8, 64, 16), 256, 0, stream>>>(pred, targ, isig);
  sobelsq_blur5_kernel<<<dim3(8, 64, 32), 256, 0, stream>>>(isig, j11, j22);
  gmaps_kernel<<<dim3(4096, 16), 256, 0, stream>>>(isig, j11, j22, g_i, g_j);
  init_acc_kernel<<<1, 1, 0, stream>>>(acc);
  loss_i_kernel<<<dim3(64, 64, 8), 256, 0, stream>>>(pred, targ, g_i, acc);
  loss_j_kernel<<<dim3(64, 64, 8), 256, 0, stream>>>(j11, j22, g_j, acc);
  finalize_kernel<<<1, 1, 0, stream>>>(acc, (float*)d_out);
}